// MambaLayer_33621003993786
// MI455X (gfx1250) — compile-verified
//
#include <hip/hip_runtime.h>

// Mamba layer for MI455X (gfx1250): B=4, L=1024, H=E=N=128.
// Pipeline:
//   WMMA-f32 GEMM(in, fused silu-z)  -> depthwise conv+silu
//   -> WMMA-f32 GEMM(ssm, fused softplus) -> 3-pass chunked selective scan
//   -> WMMA-f32 GEMM(out)

#define HDIM 128
#define BATCH 4
#define LEN 1024
#define MTOT (BATCH * LEN)   // 4096
#define NCHUNK 16
#define CLEN (LEN / NCHUNK)  // 64

typedef float v2f __attribute__((ext_vector_type(2)));
typedef float v8f __attribute__((ext_vector_type(8)));

__device__ __forceinline__ float sigmoid_fast(float v) { return 1.0f / (1.0f + __expf(-v)); }
__device__ __forceinline__ float silu_fast(float v)    { return v * sigmoid_fast(v); }
__device__ __forceinline__ float softplus_fast(float v){ return (v > 20.0f) ? v : log1pf(__expf(v)); }

// Epilogue modes
#define EPI_XZ    0  // N=256: cols [0,128) -> x_raw, cols [128,256) -> silu -> silu_z
#define EPI_SSM   1  // N=384: [0,128) softplus->delta, [128,256)->Bm, [256,384)->C
#define EPI_PLAIN 2  // plain store to out0 (ldc = HDIM)

// One wave computes one 16x16 tile of out = A(MxK) * W(KxN) + bias using
// V_WMMA_F32_16X16X4_F32. N, K, MODE are compile-time so the fully unrolled
// K-loop becomes immediate-offset global loads + 32 WMMAs.
// Fragment layouts per CDNA5 ISA 7.12.2:
//   A 16x4 (2 VGPR):  lane&15 = M row; lane-half selects K pair {0,1}/{2,3}
//   B 4x16 (2 VGPR):  lane&15 = N col; VGPR r holds K = r + 2*(lane>>4)
//   C/D 16x16 (8 VGPR): VGPR r -> row r + 8*(lane>>4), col = lane&15
template <int N, int K, int MODE>
__global__ void __launch_bounds__(256)
gemm16_wmma_f32(const float* __restrict__ A,
                const float* __restrict__ W,
                const float* __restrict__ bias,
                float* __restrict__ out0,
                float* __restrict__ out1,
                float* __restrict__ out2)
{
    const int lane = threadIdx.x & 31;
    const int wave = threadIdx.x >> 5;
    const int gw   = blockIdx.x * (blockDim.x >> 5) + wave;
    const int tilesN = N >> 4;
    const int m0 = (gw / tilesN) << 4;
    const int n0 = (gw % tilesN) << 4;
    const int hi  = lane >> 4;   // 0 or 1 (lane half)
    const int l16 = lane & 15;

    v8f acc = {0.f, 0.f, 0.f, 0.f, 0.f, 0.f, 0.f, 0.f};

    const float* Arow = A + (size_t)(m0 + l16) * K + 2 * hi; // K offset 2*hi baked in
    const float* Wcol = W + (size_t)(2 * hi) * N + n0 + l16; // K offset 2*hi baked in

#pragma unroll
    for (int k0 = 0; k0 < K; k0 += 4) {
        v2f a;
        a.x = Arow[k0 + 0];            // K = k0 + 2*hi
        a.y = Arow[k0 + 1];            // K = k0 + 2*hi + 1
        v2f b;
        b.x = Wcol[(k0 + 0) * N];      // K = k0 + 2*hi
        b.y = Wcol[(k0 + 1) * N];      // K = k0 + 2*hi + 1
        acc = __builtin_amdgcn_wmma_f32_16x16x4_f32(false, a, false, b,
                                                    (short)0, acc, false, false);
    }

    const int   col = n0 + l16;
    const float bv  = bias[col];
#pragma unroll
    for (int r = 0; r < 8; ++r) {
        const int row = m0 + r + hi * 8;
        float v = acc[r] + bv;
        if (MODE == EPI_XZ) {
            if (col < HDIM) out0[(size_t)row * HDIM + col] = v;
            else            out1[(size_t)row * HDIM + (col - HDIM)] = silu_fast(v);
        } else if (MODE == EPI_SSM) {
            if (col < 128)      out0[(size_t)row * 128 + col]         = softplus_fast(v);
            else if (col < 256) out1[(size_t)row * 128 + (col - 128)] = v;
            else                out2[(size_t)row * 128 + (col - 256)] = v;
        } else {
            out0[(size_t)row * HDIM + col] = v;
        }
    }
}

// Depthwise conv (k=3, pad=1, per-channel) along L, then SiLU.
__global__ void conv_silu_kernel(const float* __restrict__ x,
                                 const float* __restrict__ w,
                                 const float* __restrict__ cb,
                                 float* __restrict__ x2)
{
    const int idx = blockIdx.x * blockDim.x + threadIdx.x; // over MTOT*HDIM
    const int e = idx & (HDIM - 1);
    const int m = idx >> 7;
    const int l = m & (LEN - 1);
    const float xm = (l > 0)       ? x[(size_t)(m - 1) * HDIM + e] : 0.f;
    const float xc =                 x[(size_t)m * HDIM + e];
    const float xp = (l < LEN - 1) ? x[(size_t)(m + 1) * HDIM + e] : 0.f;
    const float v  = xm * w[e * 3 + 0] + xc * w[e * 3 + 1] + xp * w[e * 3 + 2] + cb[e];
    x2[(size_t)m * HDIM + e] = silu_fast(v);
}

// ---------------- Chunked selective scan ----------------
// Recurrence per element (b,n,j):  h <- exp(d_l[n]*A[j]) * h + (d_l[n]*B_l[n]) * x_l[j]
// Chunk multiplier factorizes:     prod_l exp(d_l*A_j) = exp(A_j * sum_l d_l)
// Thread layout in every pass: tid -> j = tid&127 (H index), ng = tid>>7 owns
// 16 states n = ng*16..+15 kept in registers.

// Pass A: per (batch, chunk) workgroup. Computes Bacc (chunk scan with h0=0)
// and S[n] = sum of delta over the chunk.
__global__ void __launch_bounds__(1024)
scan_passA(const float* __restrict__ delta,
           const float* __restrict__ Bm,
           const float* __restrict__ x2,
           const float* __restrict__ A_log,
           float* __restrict__ Ssum,   // [BATCH*NCHUNK*128]
           float* __restrict__ Bacc)   // [BATCH*NCHUNK*128*128]
{
    const int wg = blockIdx.x;       // 0..BATCH*NCHUNK-1
    const int b  = wg >> 4;
    const int c  = wg & (NCHUNK - 1);
    const int tid = threadIdx.x;
    const int j   = tid & 127;
    const int ng  = tid >> 7;
    const int nbase = ng << 4;

    __shared__ float s_d[128], s_B[128], s_x[128];

    const float Aj = -__expf(A_log[j]);
    float h[16], S[16];
#pragma unroll
    for (int i = 0; i < 16; ++i) { h[i] = 0.f; S[i] = 0.f; }

    const size_t base = ((size_t)b * LEN + (size_t)c * CLEN) * HDIM;
    for (int l = 0; l < CLEN; ++l) {
        const size_t moff = base + (size_t)l * HDIM;
        if (ng == 0) {
            s_d[j] = delta[moff + j];
            s_B[j] = Bm[moff + j];
            s_x[j] = x2[moff + j];
        }
        __syncthreads();
        const float xj = s_x[j];
#pragma unroll
        for (int i = 0; i < 16; ++i) {
            const float d = s_d[nbase + i];
            S[i] += d;
            const float e = __expf(d * Aj);
            h[i] = e * h[i] + (d * s_B[nbase + i]) * xj;
        }
        __syncthreads();
    }

    const size_t ob = (size_t)(b * NCHUNK + c) * 128; // n-plane base
#pragma unroll
    for (int i = 0; i < 16; ++i)
        Bacc[(ob + nbase + i) * 128 + j] = h[i];
    if (j == 0) {
#pragma unroll
        for (int i = 0; i < 16; ++i)
            Ssum[ob + nbase + i] = S[i];
    }
}

// Pass B: per batch, sequentially combine the 16 chunk carries.
// Stores h_init (state at the START of each chunk).
__global__ void __launch_bounds__(1024)
scan_passB(const float* __restrict__ Ssum,
           const float* __restrict__ Bacc,
           const float* __restrict__ A_log,
           float* __restrict__ hinit)  // [BATCH*NCHUNK*128*128]
{
    const int b   = blockIdx.x;
    const int tid = threadIdx.x;
    const int j   = tid & 127;
    const int ng  = tid >> 7;
    const int nbase = ng << 4;

    __shared__ float s_S[128];

    const float Aj = -__expf(A_log[j]);
    float h[16];
#pragma unroll
    for (int i = 0; i < 16; ++i) h[i] = 0.f;

    for (int c = 0; c < NCHUNK; ++c) {
        const size_t ob = (size_t)(b * NCHUNK + c) * 128;
        if (ng == 0) s_S[j] = Ssum[ob + j];   // j indexes n here (both 128)
        __syncthreads();
#pragma unroll
        for (int i = 0; i < 16; ++i) {
            hinit[(ob + nbase + i) * 128 + j] = h[i];          // state at chunk start
            const float P = __expf(s_S[nbase + i] * Aj);       // chunk multiplier
            h[i] = P * h[i] + Bacc[(ob + nbase + i) * 128 + j];
        }
        __syncthreads();
    }
}

// Pass C: per (batch, chunk) workgroup. Re-run chunk from h_init, emit y*silu(z).
__global__ void __launch_bounds__(1024)
scan_passC(const float* __restrict__ delta,
           const float* __restrict__ Bm,
           const float* __restrict__ Cm,
           const float* __restrict__ x2,
           const float* __restrict__ u,
           const float* __restrict__ sz,
           const float* __restrict__ A_log,
           const float* __restrict__ Dv,
           const float* __restrict__ hinit,
           float* __restrict__ ymul)
{
    const int wg = blockIdx.x;
    const int b  = wg >> 4;
    const int c  = wg & (NCHUNK - 1);
    const int tid = threadIdx.x;
    const int j   = tid & 127;
    const int ng  = tid >> 7;
    const int nbase = ng << 4;

    __shared__ float s_d[128], s_B[128], s_x[128], s_C[128], s_u[128];
    __shared__ float s_part[8][128];

    const float Aj = -__expf(A_log[j]);
    const float Dj = Dv[j];

    const size_t ob = (size_t)(b * NCHUNK + c) * 128;
    float h[16];
#pragma unroll
    for (int i = 0; i < 16; ++i) h[i] = hinit[(ob + nbase + i) * 128 + j];

    const size_t base = ((size_t)b * LEN + (size_t)c * CLEN) * HDIM;
    for (int l = 0; l < CLEN; ++l) {
        const size_t moff = base + (size_t)l * HDIM;
        if (ng == 0) {
            s_d[j] = delta[moff + j];
            s_B[j] = Bm[moff + j];
            s_x[j] = x2[moff + j];
            s_C[j] = Cm[moff + j];
            s_u[j] = u[moff + j];
        }
        __syncthreads();
        const float xj = s_x[j];
        float partial = 0.f;
#pragma unroll
        for (int i = 0; i < 16; ++i) {
            const float d = s_d[nbase + i];
            const float e = __expf(d * Aj);
            h[i] = e * h[i] + (d * s_B[nbase + i]) * xj;
            partial += h[i];
        }
        s_part[ng][j] = partial;
        __syncthreads();
        if (ng == 0) {
            float tot = 0.f;
#pragma unroll
            for (int g = 0; g < 8; ++g) tot += s_part[g][j];
            const float y = s_C[j] * tot + Dj * s_u[j];
            ymul[moff + j] = y * sz[moff + j];
        }
        // ng>0 threads cannot pass the next iteration's first barrier until
        // ng==0 (readers of s_part, writers of s_*) arrive; 2 barriers suffice.
    }
}

extern "C" void kernel_launch(void* const* d_in, const int* in_sizes, int n_in,
                              void* d_out, int out_size, void* d_ws, size_t ws_size,
                              hipStream_t stream)
{
    (void)in_sizes; (void)n_in; (void)out_size; (void)ws_size;

    const float* u      = (const float*)d_in[0];
    const float* W_in   = (const float*)d_in[1];
    const float* b_in   = (const float*)d_in[2];
    const float* conv_w = (const float*)d_in[3];
    const float* conv_b = (const float*)d_in[4];
    const float* W_x    = (const float*)d_in[5];
    const float* b_x    = (const float*)d_in[6];
    const float* A_log  = (const float*)d_in[7];
    const float* Dv     = (const float*)d_in[8];
    const float* W_out  = (const float*)d_in[9];
    const float* b_out  = (const float*)d_in[10];

    const size_t P = (size_t)MTOT * HDIM;             // 524288 floats per plane
    const size_t NSTATE = (size_t)BATCH * NCHUNK * 128 * 128; // 1048576 floats

    float* ws    = (float*)d_ws;
    float* x_raw = ws + 0 * P;
    float* s_z   = ws + 1 * P;
    float* x2    = ws + 2 * P;
    float* delta = ws + 3 * P;
    float* Bmat  = ws + 4 * P;
    float* Cmat  = ws + 5 * P;
    float* ymul  = ws + 6 * P;
    float* Ssum  = ws + 7 * P;                        // 8192 floats
    float* Bacc  = Ssum + (size_t)BATCH * NCHUNK * 128;
    float* hinit = Bacc + NSTATE;

    // GEMM1: (4096x128)@(128x256); tiles = 256*16 = 4096 waves, 8 waves/block
    gemm16_wmma_f32<256, 128, EPI_XZ><<<512, 256, 0, stream>>>(
        u, W_in, b_in, x_raw, s_z, nullptr);
    // depthwise conv + silu
    conv_silu_kernel<<<(MTOT * HDIM) / 256, 256, 0, stream>>>(x_raw, conv_w, conv_b, x2);
    // GEMM2: (4096x128)@(128x384); tiles = 256*24 = 6144 waves
    gemm16_wmma_f32<384, 128, EPI_SSM><<<768, 256, 0, stream>>>(
        x2, W_x, b_x, delta, Bmat, Cmat);
    // chunked selective scan
    scan_passA<<<BATCH * NCHUNK, 1024, 0, stream>>>(delta, Bmat, x2, A_log, Ssum, Bacc);
    scan_passB<<<BATCH, 1024, 0, stream>>>(Ssum, Bacc, A_log, hinit);
    scan_passC<<<BATCH * NCHUNK, 1024, 0, stream>>>(delta, Bmat, Cmat, x2, u, s_z,
                                                    A_log, Dv, hinit, ymul);
    // GEMM3: (4096x128)@(128x128); tiles = 256*8 = 2048 waves
    gemm16_wmma_f32<128, 128, EPI_PLAIN><<<256, 256, 0, stream>>>(
        ymul, W_out, b_out, (float*)d_out, nullptr, nullptr);
}